// RXTXAttention_82386062671911
// MI455X (gfx1250) — compile-verified
//
#include <hip/hip_runtime.h>
#include <hip/hip_bf16.h>

// ---------------------------------------------------------------------------
// RXTX attention, MI455X (gfx1250).
// Exact algebraic rewrite: scores rows have period-4 repetition, so softmax
// over T collapses to a 4-way softmax and attn@v collapses to p(4) @ Vbar(4x4)
// with Vbar = per-(b,h) column-phase mean of v.  All GEMMs are f32 WMMA
// (V_WMMA_F32_16X16X4_F32), K=32 -> 8 k-steps, N=32 -> 2 n-tiles per wave.
// ---------------------------------------------------------------------------

typedef __attribute__((ext_vector_type(2))) float v2f;
typedef __attribute__((ext_vector_type(8))) float v8f;

#define BB   4
#define TT   2048
#define CC   32
#define HH   8
#define NB   512        // TT/4
#define ROWS (BB*TT)    // 8192

__device__ __forceinline__ float clip100(float x) {
    return fminf(fmaxf(x, -100.0f), 100.0f);
}

// ---------------------------------------------------------------------------
// Kernel 1: q = x@Wq^T + bq ; v = x@Wv^T + bv   (fused, shares A-tile loads)
// One wave per 16-row M tile.  8192/16 = 512 tiles = 64 blocks * 8 waves.
// A lane layout (16x4 f32):  lane m=lane&15, kh=lane>>4; a[v]=X[m][k0+2*kh+v]
// B lane layout (4x16 f32):  lane n=lane&15 (+n0), same K split; b[v]=W[n][k0+2*kh+v]
// C/D (16x16 f32): vgpr r -> row r+8*kh, col lane&15 (+n0)
// ---------------------------------------------------------------------------
__global__ __launch_bounds__(256) void qv_wmma_kernel(
    const float* __restrict__ x,
    const float* __restrict__ Wq, const float* __restrict__ bq,
    const float* __restrict__ Wv, const float* __restrict__ bv,
    float* __restrict__ q, float* __restrict__ v)
{
    const int lane = threadIdx.x & 31;
    const int wv   = threadIdx.x >> 5;
    const int mt   = blockIdx.x * 8 + wv;      // 0..511
    const int m    = lane & 15;
    const int kh   = lane >> 4;                // 0/1 -> K pairs {0,1}/{2,3}
    const int nlo  = lane & 15;
    const int nhi  = nlo + 16;

    const float* xr = x + (mt * 16 + m) * CC;

    v8f q0 = {}, q1 = {}, v0 = {}, v1 = {};

#pragma unroll
    for (int kk = 0; kk < 8; ++kk) {
        const int k0 = kk * 4 + kh * 2;
        v2f a;   a.x   = xr[k0];               a.y   = xr[k0 + 1];
        v2f bql; bql.x = Wq[nlo * CC + k0];    bql.y = Wq[nlo * CC + k0 + 1];
        v2f bqh; bqh.x = Wq[nhi * CC + k0];    bqh.y = Wq[nhi * CC + k0 + 1];
        v2f bvl; bvl.x = Wv[nlo * CC + k0];    bvl.y = Wv[nlo * CC + k0 + 1];
        v2f bvh; bvh.x = Wv[nhi * CC + k0];    bvh.y = Wv[nhi * CC + k0 + 1];
        q0 = __builtin_amdgcn_wmma_f32_16x16x4_f32(false, a, false, bql, (short)0, q0, false, false);
        q1 = __builtin_amdgcn_wmma_f32_16x16x4_f32(false, a, false, bqh, (short)0, q1, false, false);
        v0 = __builtin_amdgcn_wmma_f32_16x16x4_f32(false, a, false, bvl, (short)0, v0, false, false);
        v1 = __builtin_amdgcn_wmma_f32_16x16x4_f32(false, a, false, bvh, (short)0, v1, false, false);
    }

    const float bq_lo = bq[nlo], bq_hi = bq[nhi];
    const float bv_lo = bv[nlo], bv_hi = bv[nhi];
#pragma unroll
    for (int r = 0; r < 8; ++r) {
        const int row = mt * 16 + kh * 8 + r;
        q[row * CC + nlo] = q0[r] + bq_lo;
        q[row * CC + nhi] = q1[r] + bq_hi;
        v[row * CC + nlo] = v0[r] + bv_lo;
        v[row * CC + nhi] = v1[r] + bv_hi;
    }
}

// ---------------------------------------------------------------------------
// Kernel 2: Vbar[b,h,c,d] = (1/512) * sum_j v[b, 4j+c, 4h+d]
// One block per (b,h); deterministic LDS tree reduction (no float atomics).
// ---------------------------------------------------------------------------
__global__ __launch_bounds__(256) void vbar_kernel(
    const float* __restrict__ v, float* __restrict__ vbar)
{
    __shared__ float red[256];
    const int bh = blockIdx.x;            // b*8+h
    const int b  = bh >> 3;
    const int h  = bh & 7;
    const int t  = threadIdx.x;
    const int e  = t & 15;                // c*4+d
    const int c  = e >> 2;
    const int d  = e & 3;

    float s = 0.0f;
    for (int j = (t >> 4); j < NB; j += 16)
        s += v[(b * TT + 4 * j + c) * CC + 4 * h + d];
    red[t] = s;
    __syncthreads();
#pragma unroll
    for (int off = 128; off >= 16; off >>= 1) {
        if (t < off) red[t] += red[t + off];
        __syncthreads();
    }
    if (t < 16) vbar[bh * 16 + t] = red[t] * (1.0f / (float)NB);
}

// ---------------------------------------------------------------------------
// Kernel 3: per 4x4 q block: RXTX -> clip -> /2 -> row softmax -> p @ Vbar
// One thread per (b,h,i) block; 4*8*512 = 16384 threads.
// ---------------------------------------------------------------------------
__global__ __launch_bounds__(256) void rxtx_softmax_kernel(
    const float* __restrict__ q,
    const float* __restrict__ vbar,
    float* __restrict__ attn_out)
{
    const int g = blockIdx.x * 256 + threadIdx.x;    // < 16384
    const int b = g >> 12;
    const int h = (g >> 9) & 7;
    const int i = g & 511;
    const int trow = b * TT + 4 * i;
    const int cb   = 4 * h;

    float X[4][4];
#pragma unroll
    for (int r = 0; r < 4; ++r) {
        float4 t = *(const float4*)(q + (trow + r) * CC + cb);
        X[r][0] = clip100(t.x); X[r][1] = clip100(t.y);
        X[r][2] = clip100(t.z); X[r][3] = clip100(t.w);
    }
    const float x1=X[0][0],  x2=X[0][1],  x3=X[0][2],  x4=X[0][3];
    const float x5=X[1][0],  x6=X[1][1],  x7=X[1][2],  x8=X[1][3];
    const float x9=X[2][0],  x10=X[2][1], x11=X[2][2], x12=X[2][3];
    const float x13=X[3][0], x14=X[3][1], x15=X[3][2], x16=X[3][3];

    const float m1  = (-x2 + x3 - x4 + x8) * (x8 + x11);
    const float m2  = (x1 - x5 - x6 + x7) * (x15 + x5);
    const float m3  = (-x2 + x12) * (-x10 + x16 + x12);
    const float m4  = (x9 - x6) * (x13 + x9 - x14);
    const float m5  = (x2 + x11) * (-x6 + x15 - x7);
    const float m6  = (x6 + x11) * (x6 + x7 - x11);
    const float m7  = x11 * (x6 + x7);
    const float m8  = x2 * (-x14 - x10 + x6 - x15 + x7 + x16 + x12);
    const float m9  = x6 * (x13 + x9 - x14 - x10 + x6 + x7 - x11);
    const float m10 = (x2 - x3 + x7 + x11 + x4 - x8) * x11;
    const float m11 = (x5 + x6 - x7) * x5;
    const float m12 = (x2 - x3 + x4) * x8;
    const float m13 = (-x1 + x5 + x6 + x3 - x7 + x11) * x15;
    const float m14 = (-x1 + x5 + x6) * (x13 + x9 + x15);
    const float m15 = (x2 + x4 - x8) * (x11 + x16 + x12);
    const float m16 = (x1 - x8) * (x9 - x16);
    const float m17 = x12 * (x10 - x12);
    const float m18 = x9 * (x13 - x14);
    const float m19 = (-x2 + x3) * (-x15 + x7 + x8);
    const float m20 = (x5 + x9 - x8) * x9;
    const float m21 = x8 * (x9 - x8 + x12);
    const float m22 = (-x6 + x7) * (x5 + x7 - x11);
    const float m23 = x1 * (x13 - x5 + x16);
    const float m24 = (-x1 + x4 + x12) * x16;
    const float m25 = (x9 + x2 + x10) * x14;
    const float m26 = (x6 + x10 + x12) * x10;

    const float z1 = m7 - m11 - m12;
    const float z2 = m1 + m12 + m21;
    const float z3 = m3 + m17 - m24;
    const float z4 = m2 + m11 + m23;
    const float z5 = m5 + m7 + m8;
    const float z6 = m4 - m18 - m20;
    const float z7 = m6 - m7 - m9;
    const float z8 = m17 + m18;

    const float c01 = m2 - m5 - z1 + m13 + m19;
    const float c02 = z2 + z3 + m15 + m16;
    const float c03 = z4 - z3 - z5 - m13;
    const float c11 = m1 + m6 - z1 + m10 + m22;
    const float c12 = z2 - z6 + z7 + m10;
    const float c13 = z4 + z6 + m14 + m16;
    const float c22 = m4 - z7 - z8 + m26;
    const float c23 = m3 + z5 + z8 + m25;

    const float Cm[4][4] = { {0.0f, c01,  c02,  c03},
                             {c01,  c11,  c12,  c13},
                             {c02,  c12,  c22,  c23},
                             {c03,  c13,  c23,  0.0f} };

    float Vb[4][4];
#pragma unroll
    for (int e = 0; e < 16; ++e)
        Vb[e >> 2][e & 3] = vbar[(b * 8 + h) * 16 + e];

#pragma unroll
    for (int r = 0; r < 4; ++r) {
        // scores = clip(C)/sqrt(4); softmax over the 4 distinct values
        float s0 = clip100(Cm[r][0]) * 0.5f;
        float s1 = clip100(Cm[r][1]) * 0.5f;
        float s2 = clip100(Cm[r][2]) * 0.5f;
        float s3 = clip100(Cm[r][3]) * 0.5f;
        float mx = fmaxf(fmaxf(s0, s1), fmaxf(s2, s3));
        float e0 = expf(s0 - mx), e1 = expf(s1 - mx);
        float e2 = expf(s2 - mx), e3 = expf(s3 - mx);
        float inv = 1.0f / (e0 + e1 + e2 + e3);
        float p0 = e0 * inv, p1 = e1 * inv, p2 = e2 * inv, p3 = e3 * inv;

        float4 o;
        o.x = p0 * Vb[0][0] + p1 * Vb[1][0] + p2 * Vb[2][0] + p3 * Vb[3][0];
        o.y = p0 * Vb[0][1] + p1 * Vb[1][1] + p2 * Vb[2][1] + p3 * Vb[3][1];
        o.z = p0 * Vb[0][2] + p1 * Vb[1][2] + p2 * Vb[2][2] + p3 * Vb[3][2];
        o.w = p0 * Vb[0][3] + p1 * Vb[1][3] + p2 * Vb[2][3] + p3 * Vb[3][3];
        *(float4*)(attn_out + (trow + r) * CC + cb) = o;
    }
}

// ---------------------------------------------------------------------------
// Kernel 4: out = attn @ Wp^T + bp   (same WMMA structure as kernel 1)
// ---------------------------------------------------------------------------
__global__ __launch_bounds__(256) void proj_wmma_kernel(
    const float* __restrict__ A,
    const float* __restrict__ Wp, const float* __restrict__ bp,
    float* __restrict__ out)
{
    const int lane = threadIdx.x & 31;
    const int wv   = threadIdx.x >> 5;
    const int mt   = blockIdx.x * 8 + wv;
    const int m    = lane & 15;
    const int kh   = lane >> 4;
    const int nlo  = lane & 15;
    const int nhi  = nlo + 16;

    const float* ar = A + (mt * 16 + m) * CC;

    v8f d0 = {}, d1 = {};
#pragma unroll
    for (int kk = 0; kk < 8; ++kk) {
        const int k0 = kk * 4 + kh * 2;
        v2f a;  a.x  = ar[k0];             a.y  = ar[k0 + 1];
        v2f bl; bl.x = Wp[nlo * CC + k0];  bl.y = Wp[nlo * CC + k0 + 1];
        v2f bh; bh.x = Wp[nhi * CC + k0];  bh.y = Wp[nhi * CC + k0 + 1];
        d0 = __builtin_amdgcn_wmma_f32_16x16x4_f32(false, a, false, bl, (short)0, d0, false, false);
        d1 = __builtin_amdgcn_wmma_f32_16x16x4_f32(false, a, false, bh, (short)0, d1, false, false);
    }

    const float b_lo = bp[nlo], b_hi = bp[nhi];
#pragma unroll
    for (int r = 0; r < 8; ++r) {
        const int row = mt * 16 + kh * 8 + r;
        out[row * CC + nlo] = d0[r] + b_lo;
        out[row * CC + nhi] = d1[r] + b_hi;
    }
}

// ---------------------------------------------------------------------------
extern "C" void kernel_launch(void* const* d_in, const int* in_sizes, int n_in,
                              void* d_out, int out_size, void* d_ws, size_t ws_size,
                              hipStream_t stream)
{
    (void)in_sizes; (void)n_in; (void)out_size; (void)ws_size;

    // setup_inputs order: x, Wq, bq, Wk, bk, Wv, bv, Wp, bp  (Wk/bk unused)
    const float* x  = (const float*)d_in[0];
    const float* Wq = (const float*)d_in[1];
    const float* bq = (const float*)d_in[2];
    const float* Wv = (const float*)d_in[5];
    const float* bv = (const float*)d_in[6];
    const float* Wp = (const float*)d_in[7];
    const float* bp = (const float*)d_in[8];
    float* out = (float*)d_out;

    float* ws   = (float*)d_ws;
    float* q    = ws;                 // 262144 floats
    float* v    = ws + 262144;        // 262144 floats
    float* attn = ws + 524288;        // 262144 floats
    float* vbar = ws + 786432;        // 512 floats

    qv_wmma_kernel     <<<64, 256, 0, stream>>>(x, Wq, bq, Wv, bv, q, v);
    vbar_kernel        <<<32, 256, 0, stream>>>(v, vbar);
    rxtx_softmax_kernel<<<64, 256, 0, stream>>>(q, vbar, attn);
    proj_wmma_kernel   <<<64, 256, 0, stream>>>(attn, Wp, bp, out);
}